// TensorfVM_61229053772018
// MI455X (gfx1250) — compile-verified
//
#include <hip/hip_runtime.h>

typedef float v2f __attribute__((ext_vector_type(2)));
typedef float v4f __attribute__((ext_vector_type(4)));

#define NN 300   // plane resolution
#define RCH 48   // channels per plane

// Precompute per-(plane,channel) line factor: s[j] = 0.5*(v[j,149] + v[j,150]).
// The line sample at (W=1, y=0) always lands on rows 149/150 with wy=0.5.
__global__ void line_factor_kernel(const float* __restrict__ vectors,
                                   float* __restrict__ s) {
    unsigned j = threadIdx.x;
    if (j < 3u * RCH) {
        v2f vv;
        __builtin_memcpy(&vv, vectors + (size_t)j * NN + 149, 8);
        s[j] = 0.5f * (vv.x + vv.y);
    }
}

// Block = (18, 16): threadIdx.x = q (plane/channel-group), threadIdx.y = local b.
// 288 threads = exactly 9 wave32s. One thread: one (b, plane), 8 channels.
// q in [0,18): p = q/6 (via compares), r0 = (q%6)*8, outputs j0..j0+7.
template<bool USE_WS>
__global__ __launch_bounds__(288) void tensorf_vm_kernel(
    const float* __restrict__ coords,
    const float* __restrict__ matrices,
    const float* __restrict__ vectors,
    const float* __restrict__ sfac,
    float* __restrict__ out,
    unsigned B)
{
    unsigned q = threadIdx.x;                       // 0..17
    unsigned b = blockIdx.x * 16u + threadIdx.y;
    if (b >= B) return;
    unsigned p  = (q >= 12u) ? 2u : ((q >= 6u) ? 1u : 0u);
    unsigned r0 = (q - p * 6u) * 8u;
    unsigned j0 = p * RCH + r0;

    // 12B coord load (global_load_b96), broadcast across the 18 q-lanes of b
    float c[3];
    __builtin_memcpy(c, coords + (size_t)b * 3u, 12);
    float x  = (p == 0u) ? c[1] : ((p == 1u) ? c[2] : c[0]);
    float y  = (p == 0u) ? c[2] : ((p == 1u) ? c[0] : c[1]);
    float xv = (p == 0u) ? c[0] : ((p == 1u) ? c[1] : c[2]);

    // ---- bilinear weights on the 300x300 plane (align_corners=False, zeros pad)
    float ix = (x + 1.0f) * 150.0f - 0.5f;
    float iy = (y + 1.0f) * 150.0f - 0.5f;
    float ix0f = floorf(ix), iy0f = floorf(iy);
    float wx = ix - ix0f,    wy = iy - iy0f;
    int ix0 = (int)ix0f, iy0 = (int)iy0f;
    int ix1 = ix0 + 1,   iy1 = iy0 + 1;
    bool vx0 = (ix0 >= 0) & (ix0 < NN);
    bool vx1 = (ix1 >= 0) & (ix1 < NN);
    bool vy0 = (iy0 >= 0) & (iy0 < NN);
    bool vy1 = (iy1 >= 0) & (iy1 < NN);
    // fold zeros-padding validity into weights; clamp indices for addressing
    float w00 = (vy0 & vx0) ? (1.0f - wy) * (1.0f - wx) : 0.0f;
    float w01 = (vy0 & vx1) ? (1.0f - wy) * wx          : 0.0f;
    float w10 = (vy1 & vx0) ? wy * (1.0f - wx)          : 0.0f;
    float w11 = (vy1 & vx1) ? wy * wx                   : 0.0f;

    // One b64 load per row covers both x-corners; select which element each
    // corner maps to (handles the clamped edge cases with zeroed weights).
    int a   = min(max(ix0, 0), NN - 2);
    int cx0 = min(max(ix0, 0), NN - 1);
    int cx1 = min(max(ix1, 0), NN - 1);
    int cy0 = min(max(iy0, 0), NN - 1);
    int cy1 = min(max(iy1, 0), NN - 1);
    bool sel0 = (cx0 != a);
    bool sel1 = (cx1 != a);
    size_t off0 = (size_t)cy0 * NN + a;
    size_t off1 = (size_t)cy1 * NN + a;

    // ---- line (vector) sample weight: W=1 so ix = xv/2; rows 149/150 fixed.
    float ixv  = xv * 0.5f;
    float iv0f = floorf(ixv);
    float wxv  = ixv - iv0f;
    int   iv0  = (int)iv0f;
    float fv = ((iv0 == 0) ? (1.0f - wxv) : 0.0f)
             + ((iv0 == -1) ? wxv : 0.0f);

    const float* mbase = matrices + (size_t)j0 * (NN * NN);

    float m[8];
#pragma unroll
    for (int k = 0; k < 8; ++k) {
        const float* ch = mbase + (size_t)k * (NN * NN);
        v2f row0, row1;
        __builtin_memcpy(&row0, ch + off0, 8);   // gather b64 (L2-resident)
        __builtin_memcpy(&row1, ch + off1, 8);   // gather b64
        float v00 = sel0 ? row0.y : row0.x;
        float v01 = sel1 ? row0.y : row0.x;
        float v10 = sel0 ? row1.y : row1.x;
        float v11 = sel1 ? row1.y : row1.x;
        m[k] = w00 * v00 + w01 * v01 + w10 * v10 + w11 * v11;
    }

    v4f o0, o1;
    if (USE_WS) {
        v4f s0, s1;
        __builtin_memcpy(&s0, sfac + j0, 16);        // hot b128, precomputed
        __builtin_memcpy(&s1, sfac + j0 + 4, 16);
#pragma unroll
        for (int k = 0; k < 4; ++k) { o0[k] = m[k] * (fv * s0[k]); }
#pragma unroll
        for (int k = 0; k < 4; ++k) { o1[k] = m[k + 4] * (fv * s1[k]); }
    } else {
        const float* vbase = vectors + (size_t)j0 * NN;
        float hf = 0.5f * fv;
#pragma unroll
        for (int k = 0; k < 8; ++k) {
            v2f vv;
            __builtin_memcpy(&vv, vbase + (size_t)k * NN + 149, 8);
            float ok = m[k] * (hf * (vv.x + vv.y));
            if (k < 4) o0[k] = ok; else o1[k - 4] = ok;
        }
    }

    // 604 MB output stream: non-temporal so it doesn't evict the 52 MB
    // matrices working set from the 192 MB L2 (TH=NT store hint on gfx1250).
    float* dst = out + (size_t)b * 144u + j0;
    __builtin_nontemporal_store(o0, (v4f*)dst);
    __builtin_nontemporal_store(o1, (v4f*)(dst + 4));
}

extern "C" void kernel_launch(void* const* d_in, const int* in_sizes, int n_in,
                              void* d_out, int out_size, void* d_ws, size_t ws_size,
                              hipStream_t stream) {
    const float* coords   = (const float*)d_in[0];  // (B, 3)
    const float* matrices = (const float*)d_in[1];  // (3, 48, 300, 300)
    const float* vectors  = (const float*)d_in[2];  // (3, 48, 300, 1)
    float* out = (float*)d_out;                     // (B, 144)

    unsigned B = (unsigned)(in_sizes[0] / 3);
    dim3 block(18, 16, 1);                          // 288 = 9 wave32s
    dim3 grid((B + 15u) / 16u, 1, 1);

    if (ws_size >= 3u * RCH * sizeof(float)) {
        float* sfac = (float*)d_ws;
        line_factor_kernel<<<1, 160, 0, stream>>>(vectors, sfac);
        tensorf_vm_kernel<true><<<grid, block, 0, stream>>>(
            coords, matrices, vectors, sfac, out, B);
    } else {
        tensorf_vm_kernel<false><<<grid, block, 0, stream>>>(
            coords, matrices, vectors, nullptr, out, B);
    }
}